// UniSphereTorchSampler_5016521801787
// MI455X (gfx1250) — compile-verified
//
#include <hip/hip_runtime.h>
#include <stdint.h>

// MI455X / gfx1250, wave32. Cosine-argmax via V_WMMA_F32_16X16X4_F32 (K=3 fits K=4),
// scatter/argmin via packed u64 GLOBAL_ATOMIC_MIN (eliminates the 256MB grid).

typedef __attribute__((ext_vector_type(2))) float v2f;
typedef __attribute__((ext_vector_type(8))) float v8f;

#define N_POINTS  65536
#define N_SAMPLES 1024
#define EMPTY_KEY 0xFFFFFFFFFFFFFFFFull
#define ROW2_OFF  (2 * N_SAMPLES + 16)   // pad row2 by 16 words: half0 banks [16t..16t+15],
                                         // half1 banks [16t+16..16t+31] -> conflict-free

// ds_swizzle_b32 xor-shuffle within 32 lanes: offset = (xor<<10) | (or<<5) | and(=0x1f)
#define SWZ_I(v, imm) __builtin_amdgcn_ds_swizzle((v), (imm))
#define SWZ_F(v, imm) __int_as_float(__builtin_amdgcn_ds_swizzle(__float_as_int(v), (imm)))

__global__ void uss_init_ws(unsigned long long* __restrict__ ws) {
    int i = blockIdx.x * blockDim.x + threadIdx.x;
    if (i < N_SAMPLES) ws[i] = EMPTY_KEY;
}

__launch_bounds__(256)
__global__ void uss_bins_kernel(const float* __restrict__ points,
                                const float* __restrict__ anchors,
                                unsigned long long* __restrict__ ws) {
    // Stage anchors (3 x 1024 f32 = 12KB) into LDS once per block, row2 padded.
    __shared__ float lds_anch[3 * N_SAMPLES + 16];
    for (int i = threadIdx.x; i < 3 * N_SAMPLES; i += blockDim.x) {
        int dst = i + ((i >= 2 * N_SAMPLES) ? 16 : 0);
        lds_anch[dst] = anchors[i];
    }
    __syncthreads();

    const int wave = threadIdx.x >> 5;        // 8 waves per block
    const int lane = threadIdx.x & 31;
    const int l    = lane & 15;               // position within half
    const int hi   = lane >> 4;               // 0 -> K=0,1 ; 1 -> K=2,3
    const int tile = blockIdx.x * 8 + wave;   // 16-point tile per wave
    const int m    = tile * 16 + l;           // this lane's point

    // Load + normalize point (both halves load the same point l; supply K-split A).
    float px = points[3 * m + 0];
    float py = points[3 * m + 1];
    float pz = points[3 * m + 2];
    float d  = sqrtf(px * px + py * py + pz * pz);
    float nrm = (d == 0.0f) ? 1.0f : d;
    float inv = 1.0f / nrm;

    // A (16x4 f32): lanes 0-15 -> {K0=x, K1=y}, lanes 16-31 -> {K2=z, K3=0}
    v2f a;
    a.x = (hi ? pz : px) * inv;
    a.y = hi ? 0.0f : (py * inv);

    const int rowx = hi ? ROW2_OFF : 0;       // which row feeds b.x for this half

    float best[8];
    int   bidx[8];
#pragma unroll
    for (int r = 0; r < 8; ++r) { best[r] = -3.402823466e38f; bidx[r] = 0; }

    // 64 anchor tiles of 16 columns each; unroll to amortize loop control
    // and overlap independent ds_load / WMMA / cndmask chains.
#pragma unroll 4
    for (int t = 0; t < N_SAMPLES / 16; ++t) {
        int s = t * 16 + l;                   // this lane's anchor column
        // B (4x16 f32): lanes 0-15 -> {K0=anch[0][s], K1=anch[1][s]},
        //               lanes 16-31 -> {K2=anch[2][s], K3=0}
        v2f b;
        b.x = lds_anch[rowx + s];
        float b1 = lds_anch[N_SAMPLES + s];   // same-address broadcast across halves
        b.y = hi ? 0.0f : b1;

        v8f c = {};
        c = __builtin_amdgcn_wmma_f32_16x16x4_f32(
                /*neg_a=*/false, a, /*neg_b=*/false, b,
                /*c_mod=*/(short)0, c, /*reuse_a=*/false, /*reuse_b=*/false);

        // c[r] = cos(point tile*16 + hi*8 + r, anchor s). Branchless per-row argmax.
        // Strict '>' + ascending s across t preserves first-max semantics.
#pragma unroll
        for (int r = 0; r < 8; ++r) {
            float v = c[r];
            bool take = v > best[r];
            best[r] = take ? v : best[r];
            bidx[r] = take ? s : bidx[r];
        }
    }

    // Branchless butterfly argmax across the 16 lanes of each half via
    // ds_swizzle xor patterns (stay within a 16-lane group on wave32).
    // Tie -> smaller anchor idx (matches jnp.argmax first-occurrence).
#define RED_STEP(imm)                                                   \
    {                                                                   \
        _Pragma("unroll")                                               \
        for (int r = 0; r < 8; ++r) {                                   \
            float ov = SWZ_F(best[r], (imm));                           \
            int   oi = SWZ_I(bidx[r], (imm));                           \
            bool take = (ov > best[r]) |                                \
                        ((ov == best[r]) & (oi < bidx[r]));             \
            best[r] = take ? ov : best[r];                              \
            bidx[r] = take ? oi : bidx[r];                              \
        }                                                               \
    }
    RED_STEP(0x041f)   // xor 1
    RED_STEP(0x081f)   // xor 2
    RED_STEP(0x101f)   // xor 4
    RED_STEP(0x201f)   // xor 8
#undef RED_STEP

    // Entry r of half `hi` is point tile*16 + hi*8 + r; the lane with l == hi*8 + r
    // already holds that point's distance d. One atomic per point, static indexing.
    unsigned long long key =
        ((unsigned long long)__float_as_uint(d) << 32) | (unsigned)m;
#pragma unroll
    for (int r = 0; r < 8; ++r) {
        if (l == hi * 8 + r) {
            atomicMin(&ws[bidx[r]], key);
        }
    }
}

__global__ void uss_finalize(const unsigned long long* __restrict__ ws,
                             long long* __restrict__ out) {
    int s = blockIdx.x * blockDim.x + threadIdx.x;
    if (s < N_SAMPLES) {
        unsigned long long k = ws[s];
        out[s] = (k == EMPTY_KEY) ? -1LL
                                  : (long long)(unsigned)(k & 0xFFFFFFFFu);
    }
}

extern "C" void kernel_launch(void* const* d_in, const int* in_sizes, int n_in,
                              void* d_out, int out_size, void* d_ws, size_t ws_size,
                              hipStream_t stream) {
    (void)in_sizes; (void)n_in; (void)out_size; (void)ws_size;
    const float* points  = (const float*)d_in[0];   // (65536, 3) f32
    const float* anchors = (const float*)d_in[1];   // (3, 1024) f32
    long long*   out     = (long long*)d_out;       // (1024,) int64
    unsigned long long* ws = (unsigned long long*)d_ws; // 1024 packed keys

    uss_init_ws<<<(N_SAMPLES + 255) / 256, 256, 0, stream>>>(ws);

    // 4096 point-tiles of 16, 8 waves (tiles) per 256-thread block -> 512 blocks.
    uss_bins_kernel<<<N_POINTS / 16 / 8, 256, 0, stream>>>(points, anchors, ws);

    uss_finalize<<<(N_SAMPLES + 255) / 256, 256, 0, stream>>>(ws, out);
}